// HarmonicResonanceLayer_75969381531988
// MI455X (gfx1250) — compile-verified
//
#include <hip/hip_runtime.h>
#include <math.h>

typedef float v2f __attribute__((ext_vector_type(2)));
typedef float v8f __attribute__((ext_vector_type(8)));

#define B_ 8
#define N_ 4096
#define H_ 256
#define F_ 6

static __device__ __forceinline__ v8f wmma_f32x4(v2f a, v2f b, v8f c) {
  // 8 args: (neg_a, A, neg_b, B, c_mod, C, reuse_a, reuse_b)
  return __builtin_amdgcn_wmma_f32_16x16x4_f32(false, a, false, b, (short)0, c,
                                               false, false);
}

// ---------------------------------------------------------------------------
// Kernel 1: fold resonators into the oscillator weights.
//   M[b,h,k] = sum_f W[f,h,k] * sin(freq_f * t[b] + phase[f,k])
// 524288 elements, one per thread. Coalesced along k.
// ---------------------------------------------------------------------------
__global__ __launch_bounds__(256) void hr_build_M(
    const float* __restrict__ t, const float* __restrict__ osc,
    const float* __restrict__ phase, float* __restrict__ M) {
  const int idx = blockIdx.x * 256 + threadIdx.x;  // < B_*H_*H_
  const int k = idx & (H_ - 1);
  const int h = (idx >> 8) & (H_ - 1);
  const int b = idx >> 16;

  const float tb = t[b];
  const float freqs[F_] = {1.f, 2.f, 4.f, 8.f, 7.f, 5.f};

  float acc = 0.f;
#pragma unroll
  for (int f = 0; f < F_; ++f) {
    const float r = sinf(fmaf(freqs[f], tb, phase[f * H_ + k]));
    acc = fmaf(osc[(f * H_ + h) * H_ + k], r, acc);
  }
  M[idx] = acc;
}

// ---------------------------------------------------------------------------
// Kernel 2: batched GEMM  out[b] = x[b] (4096x256) @ M[b] (256x256)
// using V_WMMA_F32_16X16X4_F32.
//
// Grid: (H/32 k-tiles, N/512 n-tiles, B). Block = 256 threads = 8 waves.
// Block output tile: 512 rows x 32 cols. Wave w -> rows [w*64, w*64+64),
// all 32 cols, as eight 16x16 accumulators (4 row groups x 2 col groups).
// Per K-step of 4: four global_load_b64 A-frags + one ds_load_2addr_b64
// B-frag pair feed eight v_wmma_f32_16x16x4_f32.
//
// M-slice [256h x 32k] is staged in LDS with rows (h, h+1) interleaved per k
// so a single 64-bit DS load yields the (B.x, B.y) pair each lane needs:
//   Mlds[(h>>1)*96 + k*2 + (h&1)]
// Pair stride 96 floats => the +96 offset between lane halves is 32 banks,
// keeping the two halves of the wave on disjoint banks.
// ---------------------------------------------------------------------------
__global__ __launch_bounds__(256) void hr_gemm(const float* __restrict__ x,
                                               const float* __restrict__ M,
                                               float* __restrict__ out) {
  __shared__ float Mlds[128 * 96];  // 49152 bytes

  const int kblk = blockIdx.x;  // 0..7   (32 output cols each)
  const int nblk = blockIdx.y;  // 0..7   (512 output rows each)
  const int b = blockIdx.z;     // 0..7

  const int tid = threadIdx.x;

  // ---- stage M slice [256h x 32k] into LDS (swizzled) ----
  {
    const float* src = M + ((size_t)b * H_) * H_ + kblk * 32;
#pragma unroll
    for (int i = 0; i < 32; ++i) {
      const int linear = tid + i * 256;  // 0..8191
      const int kk = linear & 31;
      const int h = linear >> 5;
      Mlds[(h >> 1) * 96 + kk * 2 + (h & 1)] = src[h * H_ + kk];
    }
  }
  __syncthreads();

  const int lane = tid & 31;
  const int wave = tid >> 5;   // 0..7 -> row group of 64
  const int lhalf = lane >> 4; // 0/1: K sub-pair for A/B frags, row+8 for C/D
  const int lmod = lane & 15;

  // A-fragment base: 16x4 f32 A layout. Lane L row = L&15,
  // cols = h0 + (L>>4)*2 .. +1  (float2).
  const float* xrow =
      x + ((size_t)b * N_ + nblk * 512 + wave * 64 + lmod) * H_ + lhalf * 2;

  // B-fragment base (float2 view of the swizzled LDS):
  // lane reads pair p = h0/2 + lhalf at column col -> float2 index p*48 + col.
  const v2f* mp = (const v2f*)Mlds;

  // acc[rg][cg]: rows +rg*16..+rg*16+15, cols +cg*16..+cg*16+15
  v8f acc[4][2];
#pragma unroll
  for (int rg = 0; rg < 4; ++rg)
#pragma unroll
    for (int cg = 0; cg < 2; ++cg) acc[rg][cg] = (v8f){};

#pragma unroll 2
  for (int h0 = 0; h0 < H_; h0 += 4) {
    v2f a[4];
#pragma unroll
    for (int rg = 0; rg < 4; ++rg)
      a[rg] = *(const v2f*)(xrow + rg * 16 * H_ + h0);

    const int p = (h0 >> 1) + lhalf;
    const v2f b0 = mp[p * 48 + lmod];
    const v2f b1 = mp[p * 48 + 16 + lmod];

#pragma unroll
    for (int rg = 0; rg < 4; ++rg) {
      acc[rg][0] = wmma_f32x4(a[rg], b0, acc[rg][0]);
      acc[rg][1] = wmma_f32x4(a[rg], b1, acc[rg][1]);
    }
  }

  // ---- epilogue: C/D layout -> VGPR r holds row r (lanes 0-15) / r+8
  // (lanes 16-31), col = lane&15. ----
  float* obase =
      out + ((size_t)b * N_ + nblk * 512 + wave * 64) * H_ + kblk * 32;
#pragma unroll
  for (int rg = 0; rg < 4; ++rg) {
#pragma unroll
    for (int r = 0; r < 8; ++r) {
      const int row = rg * 16 + lhalf * 8 + r;
      obase[row * H_ + lmod] = acc[rg][0][r];
      obase[row * H_ + 16 + lmod] = acc[rg][1][r];
    }
  }
}

extern "C" void kernel_launch(void* const* d_in, const int* in_sizes, int n_in,
                              void* d_out, int out_size, void* d_ws,
                              size_t ws_size, hipStream_t stream) {
  const float* x = (const float*)d_in[0];      // [B, N, H]
  const float* t = (const float*)d_in[1];      // [B, 1]
  const float* osc = (const float*)d_in[2];    // [F, H, H]
  const float* phase = (const float*)d_in[3];  // [F, H]
  float* out = (float*)d_out;                  // [B, N, H]
  float* M = (float*)d_ws;                     // [B, H, H] scratch (2 MB)

  (void)in_sizes;
  (void)n_in;
  (void)out_size;
  (void)ws_size;

  hr_build_M<<<(B_ * H_ * H_) / 256, 256, 0, stream>>>(t, osc, phase, M);
  hr_gemm<<<dim3(H_ / 32, N_ / 512, B_), 256, 0, stream>>>(x, M, out);
}